// MultiresolutionHashEncoder2d_45724221833330
// MI455X (gfx1250) — compile-verified
//
#include <hip/hip_runtime.h>
#include <math.h>

typedef __attribute__((ext_vector_type(2))) float v2f;
typedef __attribute__((ext_vector_type(8))) float v8f;

#define AS3 __attribute__((address_space(3)))

// ---- CDNA5 async global->LDS helpers (inline asm: portable across toolchains) ----
__device__ __forceinline__ unsigned lds_off(const void* p) {
  return (unsigned)(unsigned long long)(AS3 const void*)p;
}
__device__ __forceinline__ void async_ld_f32(unsigned ldsByteOff, const float* g) {
  // GV mode: per-lane 64-bit global address, per-lane LDS byte offset in vdst
  asm volatile("global_load_async_to_lds_b32 %0, %1, off"
               :: "v"(ldsByteOff), "v"(g) : "memory");
}
__device__ __forceinline__ void wait_async0() {
  asm volatile("s_wait_asynccnt 0" ::: "memory");
}

// =====================================================================
// Stage 1: horizontal sliding max (window kw, stride 1) of x rows.
// hbuf[(b*2+c)*512 + y][gx] = max_{j<kw} x[b][c][y][gx+j]   (unscaled; *n commutes with max)
// =====================================================================
__global__ __launch_bounds__(256) void hmax_kernel(const float* __restrict__ x,
                                                   float* __restrict__ hbuf,
                                                   int gw, int kw) {
  __shared__ float s[288];                       // 256 outputs + (kw-1) halo, kw<=32
  const int tx  = threadIdx.x;
  const int gx0 = blockIdx.x * 256;
  const size_t rowBase = ((size_t)blockIdx.z * 512 + blockIdx.y) * 512;
  const int nOut = min(256, gw - gx0);
  const int need = nOut + kw - 1;                // <= 287, gx0+need-1 <= 511
  for (int j = tx; j < need; j += 256)
    async_ld_f32(lds_off(&s[j]), x + rowBase + gx0 + j);
  wait_async0();
  __syncthreads();
  const int gx = gx0 + tx;
  if (gx < gw) {
    float m = s[tx];
    for (int j = 1; j < kw; ++j) m = fmaxf(m, s[tx + j]);
    hbuf[((size_t)blockIdx.z * 512 + blockIdx.y) * gw + gx] = m;
  }
}

// =====================================================================
// Stage 2: vertical sliding max over hbuf + scale + trunc + hash + gather.
// Writes planar features: featP[f*(8*gh*gw) + (b*gh+gy)*gw + gx]
// =====================================================================
#define S2_ROWS 47                                // 16 + kh - 1, kh<=32
__global__ __launch_bounds__(128) void vmax_hash_kernel(const float* __restrict__ hbuf,
                                                        const float* __restrict__ tbl,
                                                        float* __restrict__ featP,
                                                        int gw, int gh, int kh, float fn) {
  __shared__ float s[2 * S2_ROWS * 128];          // 48128 B
  const int tx  = threadIdx.x;
  const int gx0 = blockIdx.x * 128;
  const int gy0 = blockIdx.y * 16;
  const int b   = blockIdx.z;
  const int rows = 16 + kh - 1;
  const int gx  = gx0 + tx;
  if (gx < gw) {
    for (int c = 0; c < 2; ++c) {
      const float* src = hbuf + ((size_t)(b * 2 + c) * 512) * gw;
      for (int r = 0; r < rows; ++r) {
        int y = gy0 + r;
        if (y < 512)
          async_ld_f32(lds_off(&s[(c * S2_ROWS + r) * 128 + tx]),
                       src + (size_t)y * gw + gx);
      }
    }
  }
  wait_async0();
  __syncthreads();
  if (gx < gw) {
    const float2* t2   = (const float2*)tbl;
    const size_t plane = (size_t)8 * gh * gw;     // per-feature plane stride
    for (int jy = 0; jy < 16; ++jy) {
      int gy = gy0 + jy;
      if (gy >= gh) break;                        // uniform exit
      float m0 = s[(0 * S2_ROWS + jy) * 128 + tx];
      float m1 = s[(1 * S2_ROWS + jy) * 128 + tx];
      for (int r = 1; r < kh; ++r) {
        m0 = fmaxf(m0, s[(0 * S2_ROWS + jy + r) * 128 + tx]);
        m1 = fmaxf(m1, s[(1 * S2_ROWS + jy + r) * 128 + tx]);
      }
      unsigned u0 = (unsigned)(fn * m0);          // trunc toward zero, matches astype(uint32)
      unsigned u1 = (unsigned)(fn * m1);
      unsigned idx = (u0 ^ (u1 * 2654435761u)) & 16383u;
      float2 f = t2[idx];
      size_t o = ((size_t)b * gh + gy) * gw + gx;
      featP[o]         = f.x;
      featP[plane + o] = f.y;
    }
  }
}

// =====================================================================
// Bilinear (half-pixel centers, edge clamp) as banded GEMM via WMMA f32 16x16x4.
// Weight of input index kabs for output index o when resizing n -> 512:
// =====================================================================
__device__ __forceinline__ float bl_w(int o, int kabs, int n, float scale) {
  float sp = ((float)o + 0.5f) * scale - 0.5f;
  float fl = floorf(sp);
  int   i0 = (int)fl;
  float t  = sp - fl;
  int c0 = min(max(i0, 0), n - 1);
  int c1 = min(max(i0 + 1, 0), n - 1);
  return (c0 == kabs ? 1.0f - t : 0.0f) + (c1 == kabs ? t : 0.0f);
}

// Pass V: tmp[q][512][512] = Wv(512 x gh, banded) * featP_plane(gh x gw);  q = f*8+b
__global__ __launch_bounds__(256) void resizeV_wmma(const float* __restrict__ featP,
                                                    float* __restrict__ tmp,
                                                    int gh, int gw, int nColTiles) {
  const int lane = threadIdx.x & 31;
  // Force scalar wave id so tile/band bounds are provably uniform:
  // the K-loop then compiles to an SALU loop and EXEC stays all-1s for WMMA.
  const int wave = __builtin_amdgcn_readfirstlane(threadIdx.x >> 5);
  int tile = blockIdx.x * 8 + wave;
  const int total = 16 * 32 * nColTiles;
  if (tile >= total) return;                      // wave-uniform; EXEC stays full for WMMA
  const int ct = tile % nColTiles;
  const int rb = (tile / nColTiles) & 31;
  const int q  = tile / (nColTiles * 32);
  const float* plane  = featP + (size_t)q * gh * gw;
  float*       tplane = tmp   + (size_t)q * 512 * 512;
  const float scale = (float)gh * (1.0f / 512.0f);
  const int y0 = rb * 16;
  float syf = ((float)y0 + 0.5f) * scale - 0.5f;
  float syl = ((float)(y0 + 15) + 0.5f) * scale - 0.5f;
  const int kstart = __builtin_amdgcn_readfirstlane(min(max((int)floorf(syf), 0), gh - 1));
  const int kend   = __builtin_amdgcn_readfirstlane(min(max((int)floorf(syl) + 1, 0), gh - 1));
  const int span = kend - kstart + 1;             // <= 17, scalar
  const int row = lane & 15;                      // A: M index
  const int hi  = lane >> 4;                      // half-wave select (K0/K1 vs K2/K3 pairs)
  const int col = lane & 15;                      // B: N index
  const int gxc = min(ct * 16 + col, gw - 1);
  v8f acc = {};
  for (int kb = 0; kb < span; kb += 4) {
    int ka  = kstart + kb + (hi ? 2 : 0);
    int ka1 = ka + 1;
    v2f a, bm;
    a.x = bl_w(y0 + row, ka,  gh, scale);         // zero outside band -> exact K padding
    a.y = bl_w(y0 + row, ka1, gh, scale);
    bm.x = plane[(size_t)min(ka,  gh - 1) * gw + gxc];
    bm.y = plane[(size_t)min(ka1, gh - 1) * gw + gxc];
    acc = __builtin_amdgcn_wmma_f32_16x16x4_f32(false, a, false, bm, (short)0, acc,
                                                false, false);
  }
#pragma unroll
  for (int j = 0; j < 8; ++j) {                   // C/D layout: VGPR j -> M = j (+8 for hi half)
    int m = j + (hi ? 8 : 0);
    tplane[(size_t)(y0 + m) * 512 + ct * 16 + col] = acc[j];
  }
}

// Pass H: out[b][2i+f][512][512] = tmp(512 x gw, banded cols) * Wh^T(gw x 512)
__global__ __launch_bounds__(256) void resizeH_wmma(const float* __restrict__ tmp,
                                                    float* __restrict__ out,
                                                    int gw, int chanBase) {
  const int lane = threadIdx.x & 31;
  const int wave = __builtin_amdgcn_readfirstlane(threadIdx.x >> 5);
  int tile = blockIdx.x * 8 + wave;
  const int total = 16 * 32 * 32;
  if (tile >= total) return;
  const int ct = tile & 31;
  const int rb = (tile >> 5) & 31;
  const int q  = tile >> 10;                      // f*8+b
  const int f  = q >> 3;
  const int b  = q & 7;
  const float* tplane = tmp + (size_t)q * 512 * 512;
  const float scale = (float)gw * (1.0f / 512.0f);
  const int x0t = ct * 16;
  float sxf = ((float)x0t + 0.5f) * scale - 0.5f;
  float sxl = ((float)(x0t + 15) + 0.5f) * scale - 0.5f;
  const int kstart = __builtin_amdgcn_readfirstlane(min(max((int)floorf(sxf), 0), gw - 1));
  const int kend   = __builtin_amdgcn_readfirstlane(min(max((int)floorf(sxl) + 1, 0), gw - 1));
  const int span = kend - kstart + 1;             // scalar
  const int row = lane & 15;
  const int hi  = lane >> 4;
  const int col = lane & 15;
  const float* rowp = tplane + (size_t)(rb * 16 + row) * 512;
  v8f acc = {};
  for (int kb = 0; kb < span; kb += 4) {
    int ka = kstart + kb + (hi ? 2 : 0);
    v2f a, bm;
    a.x = rowp[min(ka,     511)];                 // clamped reads; weights are 0 past band
    a.y = rowp[min(ka + 1, 511)];
    bm.x = bl_w(x0t + col, ka,     gw, scale);
    bm.y = bl_w(x0t + col, ka + 1, gw, scale);
    acc = __builtin_amdgcn_wmma_f32_16x16x4_f32(false, a, false, bm, (short)0, acc,
                                                false, false);
  }
  float* obase = out + (((size_t)b * 32 + chanBase + f) * 512) * 512;
#pragma unroll
  for (int j = 0; j < 8; ++j) {
    int m = j + (hi ? 8 : 0);
    __builtin_nontemporal_store(acc[j],
        obase + (size_t)(rb * 16 + m) * 512 + x0t + col);  // write-once: NT keeps L2 clean
  }
}

// =====================================================================
extern "C" void kernel_launch(void* const* d_in, const int* in_sizes, int n_in,
                              void* d_out, int out_size, void* d_ws, size_t ws_size,
                              hipStream_t stream) {
  (void)in_sizes; (void)n_in; (void)out_size; (void)ws_size;
  const float* x     = (const float*)d_in[0];     // [8,2,512,512]
  const float* table = (const float*)d_in[1];     // [16,16384,2]
  float*       out   = (float*)d_out;             // [8,32,512,512]
  float* hbuf  = (float*)d_ws;                          // 16 MB: hmax grid, reused as tmp
  float* featP = hbuf + (size_t)8 * 2 * 512 * 512;      // 16 MB: planar features [2][8][gh][gw]

  const double Bb = exp((log(512.0) - log(16.0)) / 15.0);  // exact reference formula
  for (int i = 0; i < 16; ++i) {
    int n  = (int)(16.0 * pow(Bb, (double)i));
    int k  = 512 / n;
    int gh = 512 - k + 1, gw = gh;

    dim3 g1((gw + 255) / 256, 512, 16);
    hmax_kernel<<<g1, dim3(256), 0, stream>>>(x, hbuf, gw, k);

    dim3 g2((gw + 127) / 128, (gh + 15) / 16, 8);
    vmax_hash_kernel<<<g2, dim3(128), 0, stream>>>(hbuf, table + (size_t)i * 16384 * 2,
                                                   featP, gw, gh, k, (float)n);

    int nColTiles = (gw + 15) / 16;
    int tilesV = 16 * 32 * nColTiles;
    resizeV_wmma<<<dim3((tilesV + 7) / 8), dim3(256), 0, stream>>>(featP, hbuf, gh, gw,
                                                                   nColTiles);

    int tilesH = 16 * 32 * 32;
    resizeH_wmma<<<dim3((tilesH + 7) / 8), dim3(256), 0, stream>>>(hbuf, out, gw, 2 * i);
  }
}